// EqualityConstraint_16441134809748
// MI455X (gfx1250) — compile-verified
//
#include <hip/hip_runtime.h>

typedef __attribute__((ext_vector_type(2))) float v2f;
typedef __attribute__((ext_vector_type(8))) float v8f;

#define BB 2048
#define NN 256
#define MM 128
#define NT 256   // 8 waves of 32

__global__ __launch_bounds__(NT)
void eqcon_gn_kernel(const float* __restrict__ xg,
                     const float* __restrict__ pg,
                     const float* __restrict__ Ag,
                     float* __restrict__ outg) {
    __shared__ float G[MM][MM + 1];   // 128 x 129 (padded) ~64.5 KB
    __shared__ float tv[NN];          // tanh(x)
    __shared__ float dv[NN];          // 1 - tanh^2
    __shared__ float Fv[MM];          // F(x); becomes RHS during forward solve
    __shared__ float Ld[MM];          // Cholesky diagonal (sqrt)
    __shared__ float wv[MM];          // forward-solve result
    __shared__ float yv[MM];          // final solve result

    const int b    = blockIdx.x;
    const int tid  = threadIdx.x;
    const int lane = tid & 31;
    const int wave = tid >> 5;

    // ---- phase 1: t = tanh(x), d = 1 - t^2 ----
    {
        const float xv = xg[b * NN + tid];
        const float t  = tanhf(xv);
        tv[tid] = t;
        dv[tid] = 1.0f - t * t;
    }
    __syncthreads();

    // ---- phase 2a: F = A t + p  (waves 0..3; wave-uniform predicate) ----
    if (tid < MM) {
        const float* Arow = Ag + tid * NN;
        float acc = pg[b * MM + tid];
        #pragma unroll 8
        for (int j = 0; j < NN; ++j) acc += Arow[j] * tv[j];
        Fv[tid] = acc;
    }

    // ---- phase 2b: G = J J^T = A diag(d^2) A^T via fp32 WMMA 16x16x4 ----
    // 8x8 tile grid; compute upper triangle (36 tiles), mirror.
    {
        const int m16   = lane & 15;
        const int khalf = lane >> 4;          // 0: K={0,1}, 1: K={2,3}
        for (int u = wave; u < 36; u += 8) {
            int I = 0, r = u;
            while (r >= 8 - I) { r -= 8 - I; ++I; }
            const int Jt = I + r;

            const float* rowAI = Ag + (I  * 16 + m16) * NN;
            const float* rowAJ = Ag + (Jt * 16 + m16) * NN;

            v8f c = {};
            for (int k0 = 0; k0 < NN; k0 += 4) {
                const int kk = k0 + 2 * khalf;           // even -> 8B aligned
                const v2f dd = *(const v2f*)(&dv[kk]);
                const v2f av = *(const v2f*)(rowAI + kk);
                const v2f bv = *(const v2f*)(rowAJ + kk);
                const v2f a  = av * dd;                  // scale both sides by d
                const v2f bbv = bv * dd;                 // => product carries d^2
                c = __builtin_amdgcn_wmma_f32_16x16x4_f32(
                        /*neg_a=*/false, a, /*neg_b=*/false, bbv,
                        /*c_mod=*/(short)0, c, /*reuse_a=*/false, /*reuse_b=*/false);
            }
            // C/D layout: VGPR v -> M = v + 8*khalf ; N = lane&15
            const int nn = Jt * 16 + m16;
            #pragma unroll
            for (int v = 0; v < 8; ++v) {
                const int mm = I * 16 + v + 8 * khalf;
                G[mm][nn] = c[v];
                if (I != Jt) G[nn][mm] = c[v];
            }
        }
    }

    // ---- phase 3: Cholesky G = L L^T (right-looking, delayed column scale).
    // Row k (right of diag) keeps unscaled active values; column k receives
    // scaled L; diagonal sqrt kept in Ld[] so G[k][k] is never overwritten.
    for (int k = 0; k < MM; ++k) {
        __syncthreads();
        const float diag = G[k][k];
        const float dinv = 1.0f / diag;
        const float rs   = 1.0f / sqrtf(diag);
        if (tid == 0) Ld[k] = diag * rs;     // = sqrt(diag)
        const int rem = MM - 1 - k;
        // scaled L column k (read row k, write column k: disjoint addresses)
        for (int idx = tid; idx < rem; idx += NT) {
            const int i = k + 1 + idx;
            G[i][k] = G[k][i] * rs;
        }
        // symmetric trailing update: G[i][j] -= G[k][i]*G[k][j]/diag, i,j>k
        const int tj = tid & 127;
        const int ti = tid >> 7;             // 2 x 128 thread layout
        if (tj < rem) {
            const int j = k + 1 + tj;
            const float gkj = G[k][j] * dinv;
            for (int i = k + 1 + ti; i < MM; i += 2) {
                G[i][j] -= G[k][i] * gkj;
            }
        }
    }

    // ---- phase 4: forward solve L w = F ----
    for (int k = 0; k < MM; ++k) {
        __syncthreads();
        const float wk = Fv[k] / Ld[k];
        if (tid == 0) wv[k] = wk;
        for (int i = k + 1 + tid; i < MM; i += NT) {
            Fv[i] -= G[i][k] * wk;           // L[i][k] (scaled) in column k
        }
    }

    // ---- phase 5: backward solve L^T y = w ----
    for (int k = MM - 1; k >= 0; --k) {
        __syncthreads();
        const float yk = wv[k] / Ld[k];
        if (tid == 0) yv[k] = yk;
        if (tid < k) wv[tid] -= G[k][tid] * yk;   // L[k][i] (scaled), i<k
    }
    __syncthreads();

    // ---- phase 6: x_new = x - d .* (A^T y)  (coalesced column access) ----
    {
        float acc = 0.0f;
        #pragma unroll 8
        for (int i = 0; i < MM; ++i) acc += Ag[i * NN + tid] * yv[i];
        outg[b * NN + tid] = xg[b * NN + tid] - dv[tid] * acc;
    }
}

extern "C" void kernel_launch(void* const* d_in, const int* in_sizes, int n_in,
                              void* d_out, int out_size, void* d_ws, size_t ws_size,
                              hipStream_t stream) {
    (void)in_sizes; (void)n_in; (void)d_ws; (void)ws_size; (void)out_size;
    const float* x = (const float*)d_in[0];   // [2048, 256]
    const float* p = (const float*)d_in[1];   // [2048, 128]
    const float* A = (const float*)d_in[2];   // [128, 256]
    float* out = (float*)d_out;               // [2048, 256]
    eqcon_gn_kernel<<<dim3(BB), dim3(NT), 0, stream>>>(x, p, A, out);
}